// EdgeAugmentedGraphTransformer_28613072126569
// MI455X (gfx1250) — compile-verified
//
#include <hip/hip_runtime.h>
#include <math.h>

// Problem constants (match reference)
#define CB   2
#define CN   512
#define CD   64
#define CE   64
#define CH   8
#define CDD  8
#define CL   4
#define CFIN 12
#define CEPS 1e-5f

#define ROWS_TOT   (CB * CN)                 // 1024 node rows
#define PAIRS_TOT  ((size_t)CB * CN * CN)    // 524288 edge pairs
#define HM_ELEMS   (PAIRS_TOT * CH)          // 4194304
#define EDGE_ELEMS (PAIRS_TOT * CE)          // 33554432

typedef __attribute__((ext_vector_type(2))) float f32x2;
typedef __attribute__((ext_vector_type(8))) float f32x8;

__device__ __forceinline__ float wave_sum(float x) {
#pragma unroll
  for (int o = 1; o <= 16; o <<= 1) x += __shfl_xor(x, o, 32);
  return x;
}
__device__ __forceinline__ float wave_max(float x) {
#pragma unroll
  for (int o = 1; o <= 16; o <<= 1) x = fmaxf(x, __shfl_xor(x, o, 32));
  return x;
}
// reduce across the 16 lanes of each wave32 half (masks 1..8 stay in-half)
__device__ __forceinline__ float half_sum(float x) {
#pragma unroll
  for (int o = 1; o <= 8; o <<= 1) x += __shfl_xor(x, o, 32);
  return x;
}
__device__ __forceinline__ float eluf(float x) { return x > 0.f ? x : (expf(x) - 1.f); }
__device__ __forceinline__ float sigf(float x) { return 1.f / (1.f + expf(-x)); }

// ---------------------------------------------------------------- embedding
__global__ void k_embed(const float* __restrict__ atom,
                        const float* __restrict__ w1, const float* __restrict__ b1,
                        const float* __restrict__ w2, const float* __restrict__ b2,
                        float* __restrict__ nodes) {
  int r = blockIdx.x * blockDim.x + threadIdx.x;
  if (r >= ROWS_TOT) return;
  float x[CFIN];
#pragma unroll
  for (int i = 0; i < CFIN; ++i) x[i] = atom[r * CFIN + i];
  float hid[32];
#pragma unroll
  for (int j = 0; j < 32; ++j) {
    float s = b1[j];
#pragma unroll
    for (int i = 0; i < CFIN; ++i) s += x[i] * w1[i * 32 + j];
    hid[j] = fmaxf(s, 0.f);
  }
  for (int c = 0; c < CD; ++c) {
    float s = b2[c];
#pragma unroll
    for (int j = 0; j < 32; ++j) s += hid[j] * w2[j * CD + c];
    nodes[r * CD + c] = s;
  }
}

// ------------------------------------------------- node LN + Q/K/V projection
// 256 threads = 8 waves, one wave per node row
__global__ void k_ln_qkv(const float* __restrict__ nodes,
                         const float* __restrict__ g, const float* __restrict__ bta,
                         const float* __restrict__ wq, const float* __restrict__ bq,
                         const float* __restrict__ wk, const float* __restrict__ bk,
                         const float* __restrict__ wv, const float* __restrict__ bv,
                         float* __restrict__ qb, float* __restrict__ kb, float* __restrict__ vb) {
  __shared__ float nrm[8][CD];
  int wave = threadIdx.x >> 5, lane = threadIdx.x & 31;
  int row = blockIdx.x * 8 + wave;
  float x0 = nodes[row * CD + lane];
  float x1 = nodes[row * CD + lane + 32];
  float s = wave_sum(x0 + x1);
  float q = wave_sum(x0 * x0 + x1 * x1);
  float mean = s * (1.f / 64.f);
  float var = q * (1.f / 64.f) - mean * mean;
  float rstd = rsqrtf(var + CEPS);
  nrm[wave][lane]      = (x0 - mean) * rstd * g[lane]      + bta[lane];
  nrm[wave][lane + 32] = (x1 - mean) * rstd * g[lane + 32] + bta[lane + 32];
  __syncthreads();
#pragma unroll
  for (int half = 0; half < 2; ++half) {
    int c = lane + half * 32;
    float aq = bq[c], ak = bk[c], av = bv[c];
    for (int d = 0; d < CD; ++d) {
      float nd = nrm[wave][d];
      aq += nd * wq[d * CD + c];
      ak += nd * wk[d * CD + c];
      av += nd * wv[d * CD + c];
    }
    qb[row * CD + c] = aq;
    kb[row * CD + c] = ak;
    vb[row * CD + c] = av;
  }
}

// ---------------------- fused: edge LN -> gate/ebias, q·k scores -> Hm, gate
// grid = B*N*2 blocks of 256 threads (one thread per j)
__global__ void k_scores(const float* __restrict__ edges,
                         const float* __restrict__ qb, const float* __restrict__ kb,
                         const float* __restrict__ lng, const float* __restrict__ lnb,
                         const float* __restrict__ wg, const float* __restrict__ bg,
                         const float* __restrict__ we, const float* __restrict__ be,
                         float* __restrict__ Hm, float* __restrict__ gateb) {
  __shared__ float qrow[CD], wgS[CD], weS[CD], gS[CD], bS[CD];
  int p = blockIdx.x;
  int b = p >> 10;            // N*2 = 1024 blocks per batch
  int rem = p & 1023;
  int i = rem >> 1;
  int j = (rem & 1) * 256 + threadIdx.x;
  if (threadIdx.x < CD) {
    int t = threadIdx.x;
    qrow[t] = qb[((size_t)(b * CN + i)) * CD + t];
    wgS[t] = wg[t]; weS[t] = we[t]; gS[t] = lng[t]; bS[t] = lnb[t];
  }
  __syncthreads();

  size_t pair = ((size_t)(b * CN + i)) * CN + j;
  const float* ev = edges + pair * CE;
  float e[CE];
  float s = 0.f, sq = 0.f;
#pragma unroll
  for (int c = 0; c < CE; ++c) { e[c] = ev[c]; s += e[c]; sq += e[c] * e[c]; }
  float mean = s * (1.f / 64.f);
  float var = sq * (1.f / 64.f) - mean * mean;
  float rstd = rsqrtf(var + CEPS);
  float gate = bg[0], ebias = be[0];
#pragma unroll
  for (int c = 0; c < CE; ++c) {
    float t = (e[c] - mean) * rstd * gS[c] + bS[c];
    gate  += t * wgS[c];
    ebias += t * weS[c];
  }
  const float* kr = kb + ((size_t)(b * CN + j)) * CD;
  float* hmr = Hm + pair * CH;
#pragma unroll
  for (int h = 0; h < CH; ++h) {
    float sc = 0.f;
#pragma unroll
    for (int d = 0; d < CDD; ++d) sc += qrow[h * CDD + d] * kr[h * CDD + d];
    sc *= 0.35355339059327373f;                 // 1/sqrt(8)
    sc = fminf(5.f, fmaxf(-5.f, sc));
    hmr[h] = sc + ebias;
  }
  gateb[pair] = gate;
}

// ----------------------------- gated softmax over j + attn = A·v (wave/head)
// grid = B*N blocks, 256 threads = 8 waves = 8 heads
__global__ void k_attn(const float* __restrict__ Hm, const float* __restrict__ gateb,
                       const float* __restrict__ vb, float* __restrict__ attnb) {
  int h = threadIdx.x >> 5, lane = threadIdx.x & 31;
  int bi = blockIdx.x;             // b*N + i
  int b = bi >> 9;                 // N = 512
  size_t base = (size_t)bi * CN;   // pair base
  const float* grow = gateb + base;
  float hmv[16];
  float mx = -3.4e38f;
#pragma unroll
  for (int jj = 0; jj < 16; ++jj) {
    int j = lane + jj * 32;
    float v = Hm[(base + j) * CH + h];
    hmv[jj] = v;
    mx = fmaxf(mx, v);
  }
  mx = wave_max(mx);
  float S = 0.f;
#pragma unroll
  for (int jj = 0; jj < 16; ++jj) { hmv[jj] = expf(hmv[jj] - mx); S += hmv[jj]; }
  S = wave_sum(S);
  float invS = 1.f / S;
  float acc[CDD];
#pragma unroll
  for (int d = 0; d < CDD; ++d) acc[d] = 0.f;
  const float* vbase = vb + ((size_t)b * CN) * CD;
#pragma unroll
  for (int jj = 0; jj < 16; ++jj) {
    int j = lane + jj * 32;
    float wgt = sigf(grow[j]) * hmv[jj] * invS;
    const float* vr = vbase + (size_t)j * CD + h * CDD;
#pragma unroll
    for (int d = 0; d < CDD; ++d) acc[d] += wgt * vr[d];
  }
#pragma unroll
  for (int d = 0; d < CDD; ++d) acc[d] = wave_sum(acc[d]);
  if (lane == 0) {
    float* o = attnb + (size_t)bi * CD + h * CDD;
#pragma unroll
    for (int d = 0; d < CDD; ++d) o[d] = acc[d];
  }
}

// -------------------- node update: proj + residual + FFN + final LN (64 thr)
__global__ void k_node_update(const float* __restrict__ attnb, float* __restrict__ nodes,
                              const float* __restrict__ wln, const float* __restrict__ bln,
                              const float* __restrict__ fg, const float* __restrict__ fb,
                              const float* __restrict__ w1, const float* __restrict__ b1,
                              const float* __restrict__ w2, const float* __restrict__ b2,
                              const float* __restrict__ lg, const float* __restrict__ lb) {
  __shared__ float red[66];
  __shared__ float fns[CD];
  __shared__ float hs[128];
  int r = blockIdx.x, t = threadIdx.x;
  const float* ar = attnb + (size_t)r * CD;
  float y = bln[t];
  for (int d = 0; d < CD; ++d) y += ar[d] * wln[d * CD + t];
  float nh = y + nodes[(size_t)r * CD + t];

  red[t] = nh; __syncthreads();
  if (t == 0) {
    float s = 0.f, q = 0.f;
    for (int i = 0; i < CD; ++i) { s += red[i]; q += red[i] * red[i]; }
    float m = s * (1.f / 64.f);
    red[64] = m; red[65] = q * (1.f / 64.f) - m * m;
  }
  __syncthreads();
  float mean = red[64], rstd = rsqrtf(red[65] + CEPS);
  fns[t] = (nh - mean) * rstd * fg[t] + fb[t];
  __syncthreads();
  float h0 = b1[t], h1 = b1[t + 64];
  for (int d = 0; d < CD; ++d) {
    float x = fns[d];
    h0 += x * w1[d * 128 + t];
    h1 += x * w1[d * 128 + t + 64];
  }
  hs[t] = eluf(h0); hs[t + 64] = eluf(h1);
  __syncthreads();
  float o = b2[t];
  for (int d = 0; d < 128; ++d) o += hs[d] * w2[d * CD + t];
  nh += o;
  __syncthreads();
  red[t] = nh; __syncthreads();
  if (t == 0) {
    float s = 0.f, q = 0.f;
    for (int i = 0; i < CD; ++i) { s += red[i]; q += red[i] * red[i]; }
    float m = s * (1.f / 64.f);
    red[64] = m; red[65] = q * (1.f / 64.f) - m * m;
  }
  __syncthreads();
  mean = red[64]; rstd = rsqrtf(red[65] + CEPS);
  nodes[(size_t)r * CD + t] = (nh - mean) * rstd * lg[t] + lb[t];
}

// -------- edge update: Hm@wle + edges -> LN -> FFN(64->128->64) -> LN (WMMA)
// 128 threads = 4 waves; each wave owns a 16-row tile; 8 tiles per wave.
__global__ void __launch_bounds__(128, 1)
k_edge_update(const float* __restrict__ Hm,
              const float* __restrict__ ein, float* __restrict__ eout,
              const float* __restrict__ wle, const float* __restrict__ ble,
              const float* __restrict__ feg, const float* __restrict__ feb,
              const float* __restrict__ w1,  const float* __restrict__ b1,
              const float* __restrict__ w2,  const float* __restrict__ b2,
              const float* __restrict__ lgf, const float* __restrict__ lbf) {
  __shared__ float stage[4][2112];            // per-wave staging (16x132 max)
  int tid = threadIdx.x;
  int wave = tid >> 5, lane = tid & 31;
  int hf = lane >> 4, l16 = lane & 15;
  float* st = stage[wave];
  int waveGlobal = blockIdx.x * 4 + wave;     // 4096 waves total

  for (int s = 0; s < 8; ++s) {
    int tile = waveGlobal + s * 4096;         // 0..32767
    size_t rowBase = (size_t)tile * 16;

    // ---- A fragments of Hm (16x8, K in two steps of 4)
    f32x2 a0, a1;
    {
      const float* hp = Hm + (rowBase + l16) * CH + hf * 2;
      a0.x = hp[0]; a0.y = hp[1];
      a1.x = hp[4]; a1.y = hp[5];
    }

    // ---- GEMM1: e_hat = Hm @ wle + ble + edges  (16x64 tile in eh[4])
    f32x8 eh[4];
#pragma unroll
    for (int nt = 0; nt < 4; ++nt) {
      f32x8 c = {0.f, 0.f, 0.f, 0.f, 0.f, 0.f, 0.f, 0.f};
      int col = nt * 16 + l16;
      f32x2 bf;
      bf.x = wle[(hf * 2 + 0) * CE + col];
      bf.y = wle[(hf * 2 + 1) * CE + col];
      c = __builtin_amdgcn_wmma_f32_16x16x4_f32(false, a0, false, bf, (short)0, c, false, false);
      bf.x = wle[(4 + hf * 2 + 0) * CE + col];
      bf.y = wle[(4 + hf * 2 + 1) * CE + col];
      c = __builtin_amdgcn_wmma_f32_16x16x4_f32(false, a1, false, bf, (short)0, c, false, false);
      float bias = ble[col];
#pragma unroll
      for (int v = 0; v < 8; ++v) {
        int m = hf * 8 + v;
        c[v] += bias + ein[(rowBase + m) * CE + col];
      }
      eh[nt] = c;
    }

    // ---- LN(ffe_ln): per-row stats across 4 N-tiles + 16 lanes of each half
    float mean[8], rstd[8];
#pragma unroll
    for (int v = 0; v < 8; ++v) {
      float sm = 0.f, sq = 0.f;
#pragma unroll
      for (int nt = 0; nt < 4; ++nt) { float x = eh[nt][v]; sm += x; sq += x * x; }
      sm = half_sum(sm); sq = half_sum(sq);
      float m = sm * (1.f / 64.f);
      mean[v] = m;
      rstd[v] = rsqrtf(sq * (1.f / 64.f) - m * m + CEPS);
    }
    __syncthreads();
#pragma unroll
    for (int nt = 0; nt < 4; ++nt) {
      int col = nt * 16 + l16;
      float g = feg[col], bb = feb[col];
#pragma unroll
      for (int v = 0; v < 8; ++v)
        st[(hf * 8 + v) * 68 + col] = (eh[nt][v] - mean[v]) * rstd[v] * g + bb;
    }
    __syncthreads();

    // ---- GEMM2: h = fe @ w1 + b1  (16x128 in acc[8])
    f32x8 acc[8];
#pragma unroll
    for (int nt = 0; nt < 8; ++nt) {
      float bias = b1[nt * 16 + l16];
#pragma unroll
      for (int v = 0; v < 8; ++v) acc[nt][v] = bias;
    }
#pragma unroll
    for (int kk = 0; kk < 16; ++kk) {
      int kbase = kk * 4 + hf * 2;
      f32x2 a; a.x = st[l16 * 68 + kbase]; a.y = st[l16 * 68 + kbase + 1];
#pragma unroll
      for (int nt = 0; nt < 8; ++nt) {
        int col = nt * 16 + l16;
        f32x2 bf;
        bf.x = w1[(kbase + 0) * 128 + col];
        bf.y = w1[(kbase + 1) * 128 + col];
        acc[nt] = __builtin_amdgcn_wmma_f32_16x16x4_f32(false, a, false, bf, (short)0, acc[nt], false, false);
      }
    }

    // ---- ELU, stage h for GEMM3
    __syncthreads();
#pragma unroll
    for (int nt = 0; nt < 8; ++nt)
#pragma unroll
      for (int v = 0; v < 8; ++v)
        st[(hf * 8 + v) * 132 + nt * 16 + l16] = eluf(acc[nt][v]);
    __syncthreads();

    // ---- GEMM3: fe2 = h @ w2 + b2 (16x64 in acc3[4])
    f32x8 acc3[4];
#pragma unroll
    for (int nt = 0; nt < 4; ++nt) {
      float bias = b2[nt * 16 + l16];
#pragma unroll
      for (int v = 0; v < 8; ++v) acc3[nt][v] = bias;
    }
#pragma unroll
    for (int kk = 0; kk < 32; ++kk) {
      int kbase = kk * 4 + hf * 2;
      f32x2 a; a.x = st[l16 * 132 + kbase]; a.y = st[l16 * 132 + kbase + 1];
#pragma unroll
      for (int nt = 0; nt < 4; ++nt) {
        int col = nt * 16 + l16;
        f32x2 bf;
        bf.x = w2[(kbase + 0) * CE + col];
        bf.y = w2[(kbase + 1) * CE + col];
        acc3[nt] = __builtin_amdgcn_wmma_f32_16x16x4_f32(false, a, false, bf, (short)0, acc3[nt], false, false);
      }
    }

    // ---- residual + final LN (ln_ef) + store
#pragma unroll
    for (int nt = 0; nt < 4; ++nt)
#pragma unroll
      for (int v = 0; v < 8; ++v) eh[nt][v] += acc3[nt][v];
#pragma unroll
    for (int v = 0; v < 8; ++v) {
      float sm = 0.f, sq = 0.f;
#pragma unroll
      for (int nt = 0; nt < 4; ++nt) { float x = eh[nt][v]; sm += x; sq += x * x; }
      sm = half_sum(sm); sq = half_sum(sq);
      float m = sm * (1.f / 64.f);
      mean[v] = m;
      rstd[v] = rsqrtf(sq * (1.f / 64.f) - m * m + CEPS);
    }
#pragma unroll
    for (int nt = 0; nt < 4; ++nt) {
      int col = nt * 16 + l16;
      float g = lgf[col], bb = lbf[col];
#pragma unroll
      for (int v = 0; v < 8; ++v)
        eout[(rowBase + hf * 8 + v) * CE + col] = (eh[nt][v] - mean[v]) * rstd[v] * g + bb;
    }
    __syncthreads();
  }
}

// ---------------------------------------------------------------- output head
__global__ void k_out(const float* __restrict__ nodes,
                      const float* __restrict__ ow1, const float* __restrict__ ob1,
                      const float* __restrict__ ow2, const float* __restrict__ ob2,
                      float* __restrict__ out) {
  int r = blockIdx.x * blockDim.x + threadIdx.x;
  if (r >= ROWS_TOT) return;
  float s = ob2[0];
  for (int j = 0; j < 32; ++j) {
    float h = ob1[j];
    for (int d = 0; d < CD; ++d) h += nodes[(size_t)r * CD + d] * ow1[d * 32 + j];
    s += fmaxf(h, 0.f) * ow2[j];
  }
  out[r] = s;
}

// ------------------------------------------------------------------- launcher
extern "C" void kernel_launch(void* const* d_in, const int* in_sizes, int n_in,
                              void* d_out, int out_size, void* d_ws, size_t ws_size,
                              hipStream_t stream) {
  (void)in_sizes; (void)n_in; (void)out_size; (void)ws_size;
  const float* atom   = (const float*)d_in[0];
  const float* edges0 = (const float*)d_in[1];
  const float* ew1 = (const float*)d_in[2];  const float* eb1 = (const float*)d_in[3];
  const float* ew2 = (const float*)d_in[4];  const float* eb2 = (const float*)d_in[5];
  const float* ow1 = (const float*)d_in[6];  const float* ob1 = (const float*)d_in[7];
  const float* ow2 = (const float*)d_in[8];  const float* ob2 = (const float*)d_in[9];
  const float* ln_e_g = (const float*)d_in[10]; const float* ln_e_b = (const float*)d_in[11];
  const float* ln_n_g = (const float*)d_in[12]; const float* ln_n_b = (const float*)d_in[13];
  const float* wq = (const float*)d_in[14]; const float* bq = (const float*)d_in[15];
  const float* wk = (const float*)d_in[16]; const float* bk = (const float*)d_in[17];
  const float* wv = (const float*)d_in[18]; const float* bv = (const float*)d_in[19];
  const float* wg = (const float*)d_in[20]; const float* bg = (const float*)d_in[21];
  const float* we = (const float*)d_in[22]; const float* be = (const float*)d_in[23];
  const float* wln = (const float*)d_in[24]; const float* bln = (const float*)d_in[25];
  const float* wle = (const float*)d_in[26]; const float* ble = (const float*)d_in[27];
  const float* ffe_ln_g = (const float*)d_in[28]; const float* ffe_ln_b = (const float*)d_in[29];
  const float* ffe_w1 = (const float*)d_in[30]; const float* ffe_b1 = (const float*)d_in[31];
  const float* ffe_w2 = (const float*)d_in[32]; const float* ffe_b2 = (const float*)d_in[33];
  const float* ffn_ln_g = (const float*)d_in[34]; const float* ffn_ln_b = (const float*)d_in[35];
  const float* ffn_w1 = (const float*)d_in[36]; const float* ffn_b1 = (const float*)d_in[37];
  const float* ffn_w2 = (const float*)d_in[38]; const float* ffn_b2 = (const float*)d_in[39];
  const float* ln_ef_g = (const float*)d_in[40]; const float* ln_ef_b = (const float*)d_in[41];
  const float* ln_nf_g = (const float*)d_in[42]; const float* ln_nf_b = (const float*)d_in[43];

  float* out_head  = (float*)d_out;            // [B*N] (FOUT=1)
  float* edges_buf = out_head + ROWS_TOT;      // [B*N*N*E] lives in d_out

  float* ws    = (float*)d_ws;
  float* nodes = ws;                  // 65536
  float* qb    = nodes + 65536;
  float* kb    = qb + 65536;
  float* vb    = kb + 65536;
  float* attnb = vb + 65536;
  float* Hmb   = attnb + 65536;       // 4194304
  float* gateb = Hmb + HM_ELEMS;      // 524288

  k_embed<<<8, 128, 0, stream>>>(atom, ew1, eb1, ew2, eb2, nodes);

  for (int l = 0; l < CL; ++l) {
    const float* src_e = (l == 0) ? edges0 : edges_buf;
    k_ln_qkv<<<128, 256, 0, stream>>>(nodes,
        ln_n_g + l * 64, ln_n_b + l * 64,
        wq + l * 4096, bq + l * 64, wk + l * 4096, bk + l * 64,
        wv + l * 4096, bv + l * 64, qb, kb, vb);
    k_scores<<<CB * CN * 2, 256, 0, stream>>>(src_e, qb, kb,
        ln_e_g + l * 64, ln_e_b + l * 64,
        wg + l * 64, bg + l, we + l * 64, be + l, Hmb, gateb);
    k_attn<<<CB * CN, 256, 0, stream>>>(Hmb, gateb, vb, attnb);
    k_node_update<<<CB * CN, 64, 0, stream>>>(attnb, nodes,
        wln + l * 4096, bln + l * 64,
        ffn_ln_g + l * 64, ffn_ln_b + l * 64,
        ffn_w1 + l * 8192, ffn_b1 + l * 128, ffn_w2 + l * 8192, ffn_b2 + l * 64,
        ln_nf_g + l * 64, ln_nf_b + l * 64);
    k_edge_update<<<1024, 128, 0, stream>>>(Hmb, src_e, edges_buf,
        wle + l * 512, ble + l * 64,
        ffe_ln_g + l * 64, ffe_ln_b + l * 64,
        ffe_w1 + l * 8192, ffe_b1 + l * 128, ffe_w2 + l * 8192, ffe_b2 + l * 64,
        ln_ef_g + l * 64, ln_ef_b + l * 64);
  }

  k_out<<<4, 256, 0, stream>>>(nodes, ow1, ob1, ow2, ob2, out_head);
}